// det_nlx_16793322127488
// MI455X (gfx1250) — compile-verified
//
#include <hip/hip_runtime.h>

typedef float v2f __attribute__((ext_vector_type(2)));
typedef float v8f __attribute__((ext_vector_type(8)));

#define NGI  13
#define NLOC 10

// Epilogue for one 16x16 C-tile set. p1 points at nx[b0][0][g][0]; per-r
// offsets are compile-time constants (r*2*NGI*NLOC floats = 1040B, fits the
// 24-bit literal offset, so no per-row address arithmetic is emitted).
template<bool CHECK>
__device__ __forceinline__ void epilogue(
    const v8f& c11, const v8f& c12, const v8f& c21, const v8f& c22,
    const v2f nl1[5], const v2f nl2[5],
    float* __restrict__ p1, float* __restrict__ pdet,
    long long bb0, int nbatch)
{
#pragma unroll
    for (int r = 0; r < 8; ++r) {
        if (CHECK && (bb0 + r >= nbatch)) return;   // rows ascend; rest also OOB
        const float j11 = c11[r], j12 = c12[r];
        const float j21 = c21[r], j22 = c22[r];
        const float det = j11 * j22 - j21 * j12;
        // v_rcp_f32 + one Newton-Raphson step instead of IEEE div sequence
        float inv = __builtin_amdgcn_rcpf(det);
        inv = inv * __builtin_fmaf(-det, inv, 2.0f);
        const float s11 =  j11 * inv;   // invj11
        const float s12 = -j21 * inv;   // invj12
        const float s21 = -j12 * inv;   // invj21
        const float s22 =  j22 * inv;   // invj22
#pragma unroll
        for (int t = 0; t < 5; ++t) {   // v2f math -> v_pk_fma_f32 / VOPD pairs
            v2f o1 = s11 * nl1[t] + s12 * nl2[t];
            v2f o2 = s21 * nl1[t] + s22 * nl2[t];
            __builtin_nontemporal_store(o1, (v2f*)(p1 + r * 2 * NGI * NLOC + 2 * t));
            __builtin_nontemporal_store(o2, (v2f*)(p1 + r * 2 * NGI * NLOC + NGI * NLOC + 2 * t));
        }
        __builtin_nontemporal_store(fabsf(det), pdet + r * NGI);
    }
}

__global__ __launch_bounds__(256) void det_nlx_wmma_kernel(
    const float* __restrict__ x_loc,   // (B, 2, NLOC)
    const float* __restrict__ nlx,     // (2, NLOC, NGI)
    const float* __restrict__ W11,     // (NGI, NLOC)
    const float* __restrict__ W12,
    const float* __restrict__ W21,
    const float* __restrict__ W22,
    float* __restrict__ out_nx,        // (B, 2, NGI, NLOC)
    float* __restrict__ out_det,       // (B, NGI)
    int nbatch)
{
    const int lane  = threadIdx.x & 31;
    const int n     = lane & 15;       // g column (B/C N index) == A row (M)
    const int hi    = lane >> 4;       // half-wave selector
    const int kk    = hi * 2;          // K sub-offset inside a K=4 chunk
    const int wid   = blockIdx.x * (blockDim.x >> 5) + (threadIdx.x >> 5);
    const int nwav  = gridDim.x * (blockDim.x >> 5);
    const int ntiles = (nbatch + 15) >> 4;

    const v2f zero2 = {0.f, 0.f};
    const int gcl = (n < NGI) ? n : (NGI - 1);    // clamped g: loads stay in
                                                  // bounds; cols 13..15 never stored
    // ---- One-time preload: B operands B[k][g] = W[g][k], three K=4 chunks ----
    v2f b11[3], b12[3], b21[3], b22[3];
#pragma unroll
    for (int j = 0; j < 2; ++j) {                 // k = 0..3, 4..7: always valid
        const int idx = gcl * NLOC + 4 * j + kk;
        b11[j] = *(const v2f*)(W11 + idx);
        b12[j] = *(const v2f*)(W12 + idx);
        b21[j] = *(const v2f*)(W21 + idx);
        b22[j] = *(const v2f*)(W22 + idx);
    }
    {   // k = 8..11 chunk: hi lanes (k=10,11) must be TRUE zeros (they gate the
        // clamped/garbage A rows); lo lanes load k=8,9.
        const int idx = gcl * NLOC + (hi ? 0 : 8);
        v2f t11 = *(const v2f*)(W11 + idx), t12 = *(const v2f*)(W12 + idx);
        v2f t21 = *(const v2f*)(W21 + idx), t22 = *(const v2f*)(W22 + idx);
        b11[2] = hi ? zero2 : t11;  b12[2] = hi ? zero2 : t12;
        b21[2] = hi ? zero2 : t21;  b22[2] = hi ? zero2 : t22;
    }

    // ---- One-time preload: nlx^T rows for this lane's g, packed as v2f ----
    v2f nl1[5], nl2[5];
#pragma unroll
    for (int t = 0; t < 5; ++t) {
        nl1[t] = (v2f){ nlx[(2 * t) * NGI + gcl],
                        nlx[(2 * t + 1) * NGI + gcl] };
        nl2[t] = (v2f){ nlx[NLOC * NGI + (2 * t) * NGI + gcl],
                        nlx[NLOC * NGI + (2 * t + 1) * NGI + gcl] };
    }

    // ---- Grid-stride loop over 16-batch tiles ----
    for (int tile = wid; tile < ntiles; tile += nwav) {
        const long long brow  = (long long)tile * 16 + n;
        const long long browc = (brow < nbatch) ? brow : (long long)nbatch - 1; // clamp
        const float* xrow = x_loc + (size_t)browc * (2 * NLOC);

        // Prefetch next tile's rows for this wave (clamped -> always valid)
        {
            long long nxt = (long long)(tile + nwav) * 16 + n;
            if (nxt >= nbatch) nxt = nbatch - 1;
            __builtin_prefetch(x_loc + (size_t)nxt * (2 * NLOC), 0, 1);
        }

        // A operands: lane supplies A[m= n][k = k0+kk, k0+kk+1]
        v2f ax[3], ay[3];
#pragma unroll
        for (int j = 0; j < 2; ++j) {
            ax[j] = __builtin_nontemporal_load((const v2f*)(xrow + 4 * j + kk));
            ay[j] = __builtin_nontemporal_load((const v2f*)(xrow + NLOC + 4 * j + kk));
        }
        {   // k=10,11 (hi lanes): clamp address to k=0; product nulled by zero B
            const int k2 = hi ? 0 : 8;
            ax[2] = __builtin_nontemporal_load((const v2f*)(xrow + k2));
            ay[2] = __builtin_nontemporal_load((const v2f*)(xrow + NLOC + k2));
        }

        // 12 x v_wmma_f32_16x16x4_f32 : J = X * W^T accumulated over K chunks
        v8f c11 = {}, c12 = {}, c21 = {}, c22 = {};
#pragma unroll
        for (int j = 0; j < 3; ++j) {
            c11 = __builtin_amdgcn_wmma_f32_16x16x4_f32(false, ax[j], false, b11[j],
                                                        (short)0, c11, false, false);
            c12 = __builtin_amdgcn_wmma_f32_16x16x4_f32(false, ay[j], false, b12[j],
                                                        (short)0, c12, false, false);
            c21 = __builtin_amdgcn_wmma_f32_16x16x4_f32(false, ax[j], false, b21[j],
                                                        (short)0, c21, false, false);
            c22 = __builtin_amdgcn_wmma_f32_16x16x4_f32(false, ay[j], false, b22[j],
                                                        (short)0, c22, false, false);
        }

        // ---- Epilogue: det, inverse, nx rows; NT streaming stores ----
        if (n < NGI) {
            const long long b0 = (long long)tile * 16 + hi * 8;  // C: VGPR r -> M=r(+8)
            float* p1 = out_nx  + ((size_t)b0 * 2 * NGI + n) * NLOC;
            float* pd = out_det + (size_t)b0 * NGI + n;
            if ((long long)tile * 16 + 16 <= (long long)nbatch)
                epilogue<false>(c11, c12, c21, c22, nl1, nl2, p1, pd, b0, nbatch);
            else
                epilogue<true >(c11, c12, c21, c22, nl1, nl2, p1, pd, b0, nbatch);
        }
    }
}

extern "C" void kernel_launch(void* const* d_in, const int* in_sizes, int n_in,
                              void* d_out, int out_size, void* d_ws, size_t ws_size,
                              hipStream_t stream) {
    const float* x_loc = (const float*)d_in[0];
    const float* nlx   = (const float*)d_in[1];
    const float* W11   = (const float*)d_in[2];
    const float* W12   = (const float*)d_in[3];
    const float* W21   = (const float*)d_in[4];
    const float* W22   = (const float*)d_in[5];

    const int nbatch = in_sizes[0] / (2 * NLOC);

    float* out_nx  = (float*)d_out;                             // (B,2,NGI,NLOC)
    float* out_det = out_nx + (size_t)nbatch * 2 * NGI * NLOC;  // (B,NGI)

    const int ntiles        = (nbatch + 15) / 16;  // one wave32 tile = 16 batches
    const int wavesPerBlock = 8;                   // 256 threads
    const int tilesPerWave  = 8;                   // amortize preload over ~8 tiles
    int waves  = (ntiles + tilesPerWave - 1) / tilesPerWave;
    int blocks = (waves + wavesPerBlock - 1) / wavesPerBlock;
    int maxBlocks = (ntiles + wavesPerBlock - 1) / wavesPerBlock;
    if (blocks < 1) blocks = 1;
    if (blocks > maxBlocks) blocks = maxBlocks;

    det_nlx_wmma_kernel<<<blocks, 256, 0, stream>>>(
        x_loc, nlx, W11, W12, W21, W22, out_nx, out_det, nbatch);
}